// WarpingSceneFlow_47888885350996
// MI455X (gfx1250) — compile-verified
//
#include <hip/hip_runtime.h>

// WarpingSceneFlow for MI455X (gfx1250, wave32).
// Bandwidth-bound gather kernel: no GEMM structure -> no WMMA by design.
// ~100 MB working set (fits in 192 MB L2); memory floor ~5-8 us @ 23.3 TB/s.
//
// v2: 3D grid (x-tiles, y, b) kills the div/mod index decode and makes all
// batch/row-dependent math wave-uniform (SALU + s_load) -- the VALU stream
// carries only the per-lane horizontal interp, projection, and gather.

namespace {

constexpr int kB  = 8;
constexpr int kC  = 3;
constexpr int kH  = 384;
constexpr int kW  = 1280;
constexpr int kHs = 96;
constexpr int kWs = 320;
constexpr int kBlock = 256;               // 8 wave32s; kW = 5 * kBlock exactly

} // namespace

extern "C" __global__ __launch_bounds__(kBlock)
void warp_sceneflow_kernel(const float* __restrict__ image,     // B,3,H,W
                           const float* __restrict__ disp,      // B,1,Hs,Ws
                           const float* __restrict__ scene,     // B,3,Hs,Ws
                           const float* __restrict__ intrinsic, // B,3,3
                           const float* __restrict__ aug,       // B,2
                           float* __restrict__ out) {           // B,3,H,W
  const int x = blockIdx.x * kBlock + threadIdx.x;  // 0..1279 (exact tiling)
  const int y = blockIdx.y;                          // 0..383  (uniform)
  const int b = blockIdx.z;                          // 0..7    (uniform)

  const float* img_b = image + (size_t)b * kC * kH * kW;

  // gfx1250 prefetch path: warm WGP$/L2 with the near-identity gather
  // footprint (scene-flow displacement is O(10 px); cachelines overlap).
  __builtin_prefetch(img_b + (size_t)0 * kH * kW + y * kW + x, 0, 3);
  __builtin_prefetch(img_b + (size_t)1 * kH * kW + y * kW + x, 0, 3);
  __builtin_prefetch(img_b + (size_t)2 * kH * kW + y * kW + x, 0, 3);

  // Per-batch scaled intrinsics: b is uniform -> scalar loads / SALU.
  const float sy = (float)kH / aug[b * 2 + 0];
  const float sx = (float)kW / aug[b * 2 + 1];
  const float fx = intrinsic[b * 9 + 0] * sx;
  const float fy = intrinsic[b * 9 + 4] * sy;
  const float cx = intrinsic[b * 9 + 2] * sx;
  const float cy = intrinsic[b * 9 + 5] * sy;

  // Align-corners source coords shared by disp & scene upsamples.
  // Vertical part is wave-uniform (y from blockIdx).
  const float ysrc = (float)y * ((float)(kHs - 1) / (float)(kH - 1));
  int y0 = (int)floorf(ysrc); if (y0 > kHs - 1) y0 = kHs - 1;
  const int y1 = min(y0 + 1, kHs - 1);
  const float wy = ysrc - (float)y0;

  const float xsrc = (float)x * ((float)(kWs - 1) / (float)(kW - 1));
  int x0 = (int)floorf(xsrc); if (x0 > kWs - 1) x0 = kWs - 1;
  const int x1 = min(x0 + 1, kWs - 1);
  const float wx = xsrc - (float)x0;

  const float omwx = 1.0f - wx;
  const float omwy = 1.0f - wy;

  // Shared 4-tap upsample of disp + 3 scene channels (rows y0/y1 uniform).
  const float* disp_b = disp + (size_t)b * kHs * kWs;
  const float* scn_b  = scene + (size_t)b * kC * kHs * kWs;
  const int sr0 = y0 * kWs, sr1 = y1 * kWs;

  float df, s0, s1, s2;
  {
    const float* p = disp_b;
    df = omwy * (omwx * p[sr0 + x0] + wx * p[sr0 + x1]) +
         wy   * (omwx * p[sr1 + x0] + wx * p[sr1 + x1]);
    p = scn_b;
    s0 = omwy * (omwx * p[sr0 + x0] + wx * p[sr0 + x1]) +
         wy   * (omwx * p[sr1 + x0] + wx * p[sr1 + x1]);
    p = scn_b + kHs * kWs;
    s1 = omwy * (omwx * p[sr0 + x0] + wx * p[sr0 + x1]) +
         wy   * (omwx * p[sr1 + x0] + wx * p[sr1 + x1]);
    p = scn_b + 2 * kHs * kWs;
    s2 = omwy * (omwx * p[sr0 + x0] + wx * p[sr0 + x1]) +
         wy   * (omwx * p[sr1 + x0] + wx * p[sr1 + x1]);
  }

  const float disp_full = df * (float)kW;
  const float depth =
      fminf(fmaxf(fx * 0.54f / (disp_full + 1e-8f), 0.001f), 80.0f);

  // Unproject (Kinv of the affine pinhole K is closed-form), add scene flow.
  const float Xf = ((float)x - cx) / fx * depth + s0;
  const float Yf = ((float)y - cy) / fy * depth + s1;
  const float Zf = depth + s2;

  // Reproject.
  const float u  = fx * Xf + cx * Zf;
  const float v  = fy * Yf + cy * Zf;
  const float wq = Zf + 1e-8f;
  const float coordx = u / wq;
  const float coordy = v / wq;

  // Round-trip through [-1,1] normalization exactly like the reference.
  const float nx = coordx / (float)(kW - 1) * 2.0f - 1.0f;
  const float ny = coordy / (float)(kH - 1) * 2.0f - 1.0f;
  const float gx = (nx + 1.0f) * 0.5f * (float)(kW - 1);
  const float gy = (ny + 1.0f) * 0.5f * (float)(kH - 1);

  // Zeros-padding 4-tap grid sample (branchless clamp + mask).
  const float xf0 = floorf(gx), yf0 = floorf(gy);
  const float ax = gx - xf0, ay = gy - yf0;
  const float xf1 = xf0 + 1.0f, yf1 = yf0 + 1.0f;

  const float vx0 = (xf0 >= 0.0f && xf0 <= (float)(kW - 1)) ? 1.0f : 0.0f;
  const float vx1 = (xf1 >= 0.0f && xf1 <= (float)(kW - 1)) ? 1.0f : 0.0f;
  const float vy0 = (yf0 >= 0.0f && yf0 <= (float)(kH - 1)) ? 1.0f : 0.0f;
  const float vy1 = (yf1 >= 0.0f && yf1 <= (float)(kH - 1)) ? 1.0f : 0.0f;

  const int xi0 = (int)fminf(fmaxf(xf0, 0.0f), (float)(kW - 1));
  const int xi1 = (int)fminf(fmaxf(xf1, 0.0f), (float)(kW - 1));
  const int yi0 = (int)fminf(fmaxf(yf0, 0.0f), (float)(kH - 1));
  const int yi1 = (int)fminf(fmaxf(yf1, 0.0f), (float)(kH - 1));

  const int r0 = yi0 * kW, r1 = yi1 * kW;
  const float m00 = vy0 * vx0, m01 = vy0 * vx1;
  const float m10 = vy1 * vx0, m11 = vy1 * vx1;
  const float omax = 1.0f - ax, omay = 1.0f - ay;

  // Amortize the whole projective chain over all 3 channels.
  const size_t out_row = ((size_t)(b * kC) * kH + y) * kW + x;
#pragma unroll
  for (int c = 0; c < kC; ++c) {
    const float* ip = img_b + (size_t)c * kH * kW;
    const float v00 = m00 * ip[r0 + xi0];
    const float v01 = m01 * ip[r0 + xi1];
    const float v10 = m10 * ip[r1 + xi0];
    const float v11 = m11 * ip[r1 + xi1];
    out[out_row + (size_t)c * kH * kW] =
        omay * (omax * v00 + ax * v01) + ay * (omax * v10 + ax * v11);
  }
}

extern "C" void kernel_launch(void* const* d_in, const int* in_sizes, int n_in,
                              void* d_out, int out_size, void* d_ws, size_t ws_size,
                              hipStream_t stream) {
  (void)in_sizes; (void)n_in; (void)out_size; (void)d_ws; (void)ws_size;
  const float* image     = (const float*)d_in[0];
  const float* disp      = (const float*)d_in[1];
  const float* scene     = (const float*)d_in[2];
  const float* intrinsic = (const float*)d_in[3];
  const float* aug       = (const float*)d_in[4];
  float* out = (float*)d_out;

  dim3 grid(kW / kBlock, kH, kB);   // (5, 384, 8) -- exact tiling, no guard
  warp_sceneflow_kernel<<<grid, kBlock, 0, stream>>>(
      image, disp, scene, intrinsic, aug, out);
}